// flex_conv_66726611910775
// MI455X (gfx1250) — compile-verified
//
#include <hip/hip_runtime.h>

typedef float v2f __attribute__((ext_vector_type(2)));
typedef float v8f __attribute__((ext_vector_type(8)));

namespace {
constexpr int kB  = 8;     // batch
constexpr int kC  = 64;    // input channels
constexpr int kH  = 64;    // height
constexpr int kW  = 512;   // width
constexpr int kO  = 64;    // output channels
constexpr int kTW = 30;    // output columns per stripe
constexpr int kNS = 18;    // ceil(512/30) stripes
constexpr int kHB = 16;    // output rows per workgroup
constexpr int kNH = 4;     // h segments (4*16 = 64 rows)
constexpr int kODL = 128;  // GEMM rows per workgroup: 4 d * 32 o
constexpr int kXI = 68;    // Xt pitch ([px][i] layout): conflict-free b64 B-frag reads
constexpr int kUP = 34;    // U pitch (even -> aligned float2 stencil windows)
}

// out[b,o,h,w] = Box3x3(vbar)[o,h,w] + sum_{d<3} points_d(h,w) * Box3x3(u_d)[o,h,w]
//   u_d[o,p]  = sum_i weight[o,i,d] * x[i,p]            (WMMA f32 16x16x4 GEMM)
//   vbar[o,q] = u_3[o,q] - sum_{d<3} points_d(q)*u_d[o,q]
__global__ __launch_bounds__(256)
void flexconv_wmma(const float* __restrict__ x,
                   const float* __restrict__ pts,
                   const float* __restrict__ wgt,
                   float* __restrict__ out)
{
    __shared__ float Xt[32 * kXI];          // staged x row tile   [px][i]
    __shared__ float U[3][kODL * kUP];      // rolling u rows      [slot][od][px]

    const int stripe = blockIdx.x;
    const int hseg   = blockIdx.y >> 1;
    const int ohalf  = blockIdx.y & 1;
    const int b      = blockIdx.z;

    const int w0    = stripe * kTW;         // first output column of stripe
    const int wb    = w0 - 1;               // halo base column
    const int h0    = hseg * kHB;
    const int obase = ohalf * 32;

    const int tid   = threadIdx.x;
    const int wid   = tid >> 5;             // wave id 0..7 == GEMM M-tile
    const int lane  = tid & 31;
    const int m     = lane & 15;
    const int khalf = lane >> 4;

    // ---- A fragments (weight) in registers; wave owns od rows [wid*16, wid*16+16) ----
    // A layout (32-bit 16x4): lane m = M row; half-wave selects K pair; VGPR j = K within pair.
    const int odA = wid * 16 + m;           // local od row this lane supplies
    const int oA  = obase + (odA & 31);
    const int dA  = odA >> 5;
    float a0[16], a1[16];
#pragma unroll
    for (int kk = 0; kk < 16; ++kk) {
        const int i0 = 4 * kk + 2 * khalf;
        a0[kk] = wgt[oA * (kC * 4) + (i0 + 0) * 4 + dA];
        a1[kk] = wgt[oA * (kC * 4) + (i0 + 1) * 4 + dA];
    }

    auto compute_u_row = [&](int r, int s) {
        __syncthreads();                    // prior readers of Xt / slot s finished
        // ---- stage x[b, :, r, wb..wb+31] into LDS transposed [px][i] (zero outside) ----
        {
            const int px = tid & 31;
            const int w  = wb + px;
            const bool ok = (r >= 0) & (r < kH) & (w >= 0) & (w < kW);
            const bool pk = ok & (r + 1 < kH);
#pragma unroll
            for (int ii = 0; ii < 8; ++ii) {
                const int i = ((tid >> 5) << 3) + ii;
                const int gi = ((b * kC + i) * kH + r) * kW + w;
                float v = 0.0f;
                if (ok) v = x[gi];
                if (pk) __builtin_prefetch(&x[gi + kW], 0, 1);  // next row -> global_prefetch
                Xt[px * kXI + i] = v;
            }
        }
        __syncthreads();

        // ---- GEMM: u[od, px] = sum_i W2[od,i] * Xt[i,px]; K=64 in 16 steps of 4 ----
        v8f acc0 = {0.f,0.f,0.f,0.f,0.f,0.f,0.f,0.f};
        v8f acc1 = acc0;
#pragma unroll
        for (int kk = 0; kk < 16; ++kk) {
            const int i0 = 4 * kk + 2 * khalf;             // even -> aligned b64
            v2f av; av.x = a0[kk]; av.y = a1[kk];
            // B layout: lane m = N column, half-wave = K pair; i contiguous in Xt
            const v2f bv0 = *reinterpret_cast<const v2f*>(&Xt[(m)      * kXI + i0]);
            const v2f bv1 = *reinterpret_cast<const v2f*>(&Xt[(16 + m) * kXI + i0]);
            acc0 = __builtin_amdgcn_wmma_f32_16x16x4_f32(false, av, false, bv0,
                                                         (short)0, acc0, false, false);
            acc1 = __builtin_amdgcn_wmma_f32_16x16x4_f32(false, av, false, bv1,
                                                         (short)0, acc1, false, false);
        }

        // ---- scatter accumulators to U[s]: D layout lane m = N, VGPR v -> M = v + 8*khalf
#pragma unroll
        for (int v = 0; v < 8; ++v) {
            const int od = wid * 16 + v + 8 * khalf;
            U[s][od * kUP + m]      = acc0[v];
            U[s][od * kUP + 16 + m] = acc1[v];
        }
        __syncthreads();

        // ---- fold u3 -> vbar = u3 - sum_{d<3} pts_d(q)*u_d, in place on d=3 rows ----
#pragma unroll
        for (int rep = 0; rep < 4; ++rep) {
            const int item = tid + rep * 256;   // 32 oo x 32 px items
            const int oo = item >> 5;
            const int px = item & 31;
            const int w  = wb + px;
            float p0 = 0.f, p1 = 0.f, p2 = 0.f;
            if ((r >= 0) & (r < kH) & (w >= 0) & (w < kW)) {
                p0 = pts[((b * 3 + 0) * kH + r) * kW + w];
                p1 = pts[((b * 3 + 1) * kH + r) * kW + w];
                p2 = pts[((b * 3 + 2) * kH + r) * kW + w];
            }
            const float u0 = U[s][(oo)      * kUP + px];
            const float u1 = U[s][(32 + oo) * kUP + px];
            const float u2 = U[s][(64 + oo) * kUP + px];
            const float u3 = U[s][(96 + oo) * kUP + px];
            U[s][(96 + oo) * kUP + px] = u3 - (p0 * u0 + p1 * u1 + p2 * u2);
        }
        __syncthreads();
    };

    // prime the rolling buffer with rows h0-1 and h0
    compute_u_row(h0 - 1, 0);
    compute_u_row(h0,     1);

    for (int h = h0; h < h0 + kHB; ++h) {
        compute_u_row(h + 1, (h + 2 - h0) % 3);   // slots now hold rows h-1, h, h+1

        // ---- 3x3 box stencil via 6-wide sliding windows (aligned float2 LDS reads) ----
        const int oo = tid >> 3;                  // 0..31
        const int q0 = (tid & 7) * 4;             // first halo column of this thread's window
        float win[4][6];
#pragma unroll
        for (int rowd = 0; rowd < 4; ++rowd)
#pragma unroll
            for (int j = 0; j < 6; ++j) win[rowd][j] = 0.f;

#pragma unroll
        for (int sl = 0; sl < 3; ++sl) {          // slot order irrelevant (row-symmetric box)
#pragma unroll
            for (int rowd = 0; rowd < 4; ++rowd) {   // u0,u1,u2,vbar
                const int base = (rowd * 32 + oo) * kUP + q0;   // even -> aligned b64
                const v2f* p2 = reinterpret_cast<const v2f*>(&U[sl][base]);
#pragma unroll
                for (int j = 0; j < 3; ++j) {
                    const v2f v = p2[j];
                    win[rowd][2 * j]     += v.x;
                    win[rowd][2 * j + 1] += v.y;
                }
            }
        }

#pragma unroll
        for (int jj = 0; jj < 4; ++jj) {
            const int pxo = q0 + jj;              // output column within stripe
            const int w   = w0 + pxo;
            if (pxo >= kTW || w >= kW) continue;
            const float su0 = win[0][jj] + win[0][jj + 1] + win[0][jj + 2];
            const float su1 = win[1][jj] + win[1][jj + 1] + win[1][jj + 2];
            const float su2 = win[2][jj] + win[2][jj + 1] + win[2][jj + 2];
            float acc       = win[3][jj] + win[3][jj + 1] + win[3][jj + 2];
            const float pc0 = pts[((b * 3 + 0) * kH + h) * kW + w];
            const float pc1 = pts[((b * 3 + 1) * kH + h) * kW + w];
            const float pc2 = pts[((b * 3 + 2) * kH + h) * kW + w];
            acc += pc0 * su0 + pc1 * su1 + pc2 * su2;
            out[((b * kO + obase + oo) * kH + h) * kW + w] = acc;
        }
        // next compute_u_row() starts with a barrier before reusing Xt / the old slot
    }
}

extern "C" void kernel_launch(void* const* d_in, const int* in_sizes, int n_in,
                              void* d_out, int out_size, void* d_ws, size_t ws_size,
                              hipStream_t stream) {
    const float* x   = (const float*)d_in[0];   // [8,64,64,512] f32
    const float* pts = (const float*)d_in[1];   // [8,3,64,512]  f32
    const float* wgt = (const float*)d_in[2];   // [64,64,4]     f32
    float* out = (float*)d_out;                 // [8,64,64,512] f32

    dim3 grid(kNS, kNH * 2, kB);                // stripes x (hseg*ohalf) x batch = 1152 WGs
    flexconv_wmma<<<grid, 256, 0, stream>>>(x, pts, wgt, out);

    (void)in_sizes; (void)n_in; (void)out_size; (void)d_ws; (void)ws_size;
}